// FLOSSAISLoss_51067161149784
// MI455X (gfx1250) — compile-verified
//
#include <hip/hip_runtime.h>
#include <hip/hip_bf16.h>
#include <math.h>

// ============================================================================
// FLOSS AIS loss for MI455X (gfx1250).
//
// Roofline: 144 MiB of reads -> ~6.2 us at 23.3 TB/s; fully bandwidth-bound.
// Strategy: one streaming pass stages 16 rows (2 batches x {4 pred, 4 target})
// per block into LDS with coalesced b128 loads, then V_WMMA_F32_16X16X4_F32
// accumulates the 16x16 Gram matrix (cross / p2 / t2 all fall out of the two
// diagonal 8x8 blocks). Consistency term computed from the same staged data.
// Deterministic: per-block partials written (no atomics), fixed-order reduces.
//
// WMMA layout assumption (compile-only, cannot verify on HW): f32 A-operand
// per ISA = lane (l&15) owns row, k = 2*(l>>4) + vgpr; B-operand assumed to
// mirror it (k = 2*(l>>4) + vgpr, col = l&15), which makes A == B for a Gram.
// ============================================================================

#define K_        4
#define NPOS      524288        // C*T elements per (b,k) row (contiguous)
#define CHUNK     512           // staged positions per block iteration
#define STRIDE    516           // LDS row stride in floats (pad 4 => no bank conflicts)
#define NBPG      128           // blocks per batch-pair group
#define NGROUP    4             // 8 batches -> 4 pairs
#define NBLOCKS   (NBPG * NGROUP)
#define PART_STRIDE 64          // floats per per-block partial record

typedef __attribute__((ext_vector_type(2))) float v2f;
typedef __attribute__((ext_vector_type(4))) float f4v;
typedef __attribute__((ext_vector_type(8))) float v8f;

__global__ __launch_bounds__(512)
void floss_main(const float* __restrict__ sep, const float* __restrict__ srcs,
                const float* __restrict__ mix, float* __restrict__ part)
{
    __shared__ float smem[16 * STRIDE];          // 33 KiB staging buffer
    const int tid  = threadIdx.x;
    const int w    = tid >> 5;                   // wave id (wave32)
    const int lane = tid & 31;
    const int g    = blockIdx.x / NBPG;          // batch-pair group
    const int rb   = blockIdx.x % NBPG;

    // Load phase: wave w streams row w. Rows: [pred b0 | tgt b0 | pred b1 | tgt b1]
    const int p  = w >> 3;                       // batch within pair
    const int sk = (w >> 2) & 1;                 // 0 = separated (pred), 1 = sources (target)
    const int si = w & 3;                        // source index
    const float* rowp = (sk ? srcs : sep) + ((size_t)((2 * g + p) * K_ + si)) * NPOS;
    const float* mix0 = mix + (size_t)(2 * g)     * NPOS;
    const float* mix1 = mix + (size_t)(2 * g + 1) * NPOS;

    // WMMA operand ownership for this lane (f32 16x16x4 A layout per ISA).
    const int r   = lane & 15;                   // matrix row owned by this lane
    const int kb  = (lane >> 4) * 2;             // k-base owned by this lane
    const int sub = w * 32;                      // 32 positions per wave per chunk

    v8f acc0 = {};
    v8f acc1 = {};
    float consLoc = 0.f;

    for (int chunk = rb; chunk < NPOS / CHUNK; chunk += NBPG) {
        const int base = chunk * CHUNK;

        // ---- coalesced global -> LDS staging (b128 per lane) ----
        #pragma unroll
        for (int it = 0; it < 4; ++it) {
            const int off = it * 128 + lane * 4;
            *(f4v*)&smem[w * STRIDE + off] = *(const f4v*)(rowp + base + off);
        }
        __syncthreads();

        // ---- Gram accumulation: 8 WMMAs cover this wave's 32 positions ----
        #pragma unroll
        for (int s8 = 0; s8 < 8; ++s8) {
            const v2f a = *(const v2f*)&smem[r * STRIDE + sub + s8 * 4 + kb];
            if (s8 & 1)
                acc1 = __builtin_amdgcn_wmma_f32_16x16x4_f32(
                    false, a, false, a, (short)0, acc1, false, false);
            else
                acc0 = __builtin_amdgcn_wmma_f32_16x16x4_f32(
                    false, a, false, a, (short)0, acc0, false, false);
        }

        // ---- consistency term: sum_k sep - mixture, from staged rows ----
        {
            const float s0 = smem[0 * STRIDE + tid] + smem[1 * STRIDE + tid] +
                             smem[2 * STRIDE + tid] + smem[3 * STRIDE + tid];
            const float s1 = smem[8 * STRIDE + tid] + smem[9 * STRIDE + tid] +
                             smem[10 * STRIDE + tid] + smem[11 * STRIDE + tid];
            const float d0 = s0 - mix0[base + tid];
            const float d1 = s1 - mix1[base + tid];
            consLoc += d0 * d0 + d1 * d1;
        }
        __syncthreads();
    }

    // ---- block reduction: dump each wave's D tile, reduce in fixed order ----
    const int half = lane >> 4, col = lane & 15;
    #pragma unroll
    for (int v = 0; v < 8; ++v) {
        const int m = v + 8 * half;              // D row per ISA C/D layout
        smem[w * 256 + m * 16 + col] = acc0[v] + acc1[v];
    }
    smem[4096 + tid] = consLoc;
    __syncthreads();

    float dsum = 0.f;
    if (tid < 256) {
        #pragma unroll
        for (int w2 = 0; w2 < 16; ++w2) dsum += smem[w2 * 256 + tid];
    }
    for (int st = 256; st > 0; st >>= 1) {       // deterministic tree reduce (cons)
        if (tid < st) smem[4096 + tid] += smem[4096 + tid + st];
        __syncthreads();
    }

    // ---- extract useful Gram entries -> per-block partial record ----
    if (tid < 256) {
        const int m = tid >> 4, nn = tid & 15;
        const int pd = m >> 3;
        int slot = -1;
        if ((nn >> 3) == pd) {                   // diagonal 8x8 block only
            const int mm = m & 7, cn = nn & 7;
            if (mm < 4) {
                if (cn >= 4)       slot = pd * 16 + mm * 4 + (cn - 4);  // cross[i][j]
                else if (cn == mm) slot = 32 + pd * 4 + mm;             // p2[i]
            } else if (cn == mm)   slot = 40 + pd * 4 + (mm - 4);       // t2[j]
        }
        if (slot >= 0) part[(size_t)blockIdx.x * PART_STRIDE + slot] = dsum;
    }
    if (tid == 0) part[(size_t)blockIdx.x * PART_STRIDE + 48] = smem[4096];
}

__global__ __launch_bounds__(256)
void floss_final(const float* __restrict__ part, const float* __restrict__ pred_bits,
                 const int* __restrict__ target_bits, const float* __restrict__ flow_loss,
                 float* __restrict__ out)
{
    __shared__ float sCross[128];
    __shared__ float sP2[32];
    __shared__ float sT2[32];
    __shared__ float sCons;
    __shared__ float sBce[256];
    const int t = threadIdx.x;

    if (t < 128) {                               // cross[b][i][j]
        const int b = t >> 4, i = (t >> 2) & 3, j = t & 3;
        const int g = b >> 1, p = b & 1;
        const int slot = p * 16 + i * 4 + j;
        float s = 0.f;
        for (int rr = 0; rr < NBPG; ++rr) s += part[(size_t)(g * NBPG + rr) * PART_STRIDE + slot];
        sCross[t] = s;
    } else if (t < 160) {                        // p2[b][i]
        const int q = t - 128, b = q >> 2, i = q & 3;
        const int g = b >> 1, p = b & 1;
        const int slot = 32 + p * 4 + i;
        float s = 0.f;
        for (int rr = 0; rr < NBPG; ++rr) s += part[(size_t)(g * NBPG + rr) * PART_STRIDE + slot];
        sP2[q] = s;
    } else if (t < 192) {                        // t2[b][j]
        const int q = t - 160, b = q >> 2, j = q & 3;
        const int g = b >> 1, p = b & 1;
        const int slot = 40 + p * 4 + j;
        float s = 0.f;
        for (int rr = 0; rr < NBPG; ++rr) s += part[(size_t)(g * NBPG + rr) * PART_STRIDE + slot];
        sT2[q] = s;
    } else if (t == 192) {                       // consistency sum over all blocks
        float s = 0.f;
        for (int bi = 0; bi < NBLOCKS; ++bi) s += part[(size_t)bi * PART_STRIDE + 48];
        sCons = s;
    }

    // BCE over 8*4*64 = 2048 bits, 8 per thread
    float acc = 0.f;
    #pragma unroll
    for (int e = 0; e < 8; ++e) {
        const int idx = t * 8 + e;
        float pb = pred_bits[idx];
        pb = fminf(fmaxf(pb, 1e-7f), 1.f - 1e-7f);
        const float tb = (float)target_bits[idx];
        acc -= tb * logf(pb) + (1.f - tb) * logf(1.f - pb);
    }
    sBce[t] = acc;
    __syncthreads();
    for (int st = 128; st > 0; st >>= 1) {
        if (t < st) sBce[t] += sBce[t + st];
        __syncthreads();
    }

    if (t == 0) {
        const float invN = 1.f / (float)NPOS;
        float reconSum = 0.f;
        for (int b = 0; b < 8; ++b) {
            float cost[4][4];
            for (int i = 0; i < 4; ++i)
                for (int j = 0; j < 4; ++j)
                    cost[i][j] = (sP2[b * 4 + i] + sT2[b * 4 + j]
                                  - 2.f * sCross[b * 16 + i * 4 + j]) * invN;
            bool used[4] = {false, false, false, false};
            for (int i = 0; i < 4; ++i) {        // greedy, first-min tie-break like argmin
                float best = INFINITY; int bj = 0;
                for (int j = 0; j < 4; ++j)
                    if (!used[j] && cost[i][j] < best) { best = cost[i][j]; bj = j; }
                used[bj] = true;
                reconSum += best;
            }
        }
        const float flow  = flow_loss[0] * 1.0f;
        const float recon = reconSum * (1.f / 8.f) * 0.1f;
        const float cons  = sCons * (1.f / 4194304.f) * 0.1f;   // mean over B*C*T
        const float bit   = sBce[0] * (1.f / 2048.f) * 0.5f;
        out[0] = flow; out[1] = recon; out[2] = cons; out[3] = bit;
        out[4] = flow + recon + cons + bit;
    }
}

extern "C" void kernel_launch(void* const* d_in, const int* in_sizes, int n_in,
                              void* d_out, int out_size, void* d_ws, size_t ws_size,
                              hipStream_t stream) {
    const float* separated   = (const float*)d_in[0];
    const float* sources     = (const float*)d_in[1];
    const float* mixture     = (const float*)d_in[2];
    const float* pred_bits   = (const float*)d_in[3];
    const float* flow_loss   = (const float*)d_in[4];
    const int*   target_bits = (const int*)d_in[5];
    float* part = (float*)d_ws;   // NBLOCKS * 64 floats = 128 KiB, fully rewritten each call
    float* out  = (float*)d_out;

    floss_main<<<NBLOCKS, 512, 0, stream>>>(separated, sources, mixture, part);
    floss_final<<<1, 256, 0, stream>>>(part, pred_bits, target_bits, flow_loss, out);

    (void)in_sizes; (void)n_in; (void)out_size; (void)ws_size;
}